// DistillationLoss_29386166239348
// MI455X (gfx1250) — compile-verified
//
#include <hip/hip_runtime.h>
#include <hip/hip_bf16.h>
#include <float.h>

#define B_ 4
#define T_ 1024
#define V_ 32000
#define ANS 512
#define TOPK 50
#define IGNORE_ (-100)
#define SINK_T_ 2.0f
#define SINK_EPS_ 0.5f
#define SINK_ITERS_ 10
#define CE_W_ 1.0f
#define KD_W_ 1.0f

typedef __attribute__((ext_vector_type(2))) float v2f;
typedef __attribute__((ext_vector_type(8))) float v8f;

// ---- workspace layout (units: floats) ----
#define OFF_SS     ((size_t)0)                        // [2][B][2] ints (start,size)
#define OFF_STATS  ((size_t)16)                       // [2][B][ANS][3] = inv_std, m, inv_sumexp
#define OFF_CSUM   (OFF_STATS + (size_t)2*B_*ANS*3)   // [2][V] column sums (mutated by topk)
#define OFF_TOPI   (OFF_CSUM + (size_t)2*V_)          // [2][TOPK] ints
#define OFF_Q      (OFF_TOPI + (size_t)2*TOPK)        // [2][B][ANS][TOPK]
#define OFF_W      (OFF_Q + (size_t)2*B_*ANS*TOPK)    // [B][ANS][ANS]
#define OFF_P      (OFF_W + (size_t)B_*ANS*ANS)       // [B][ANS][ANS]
#define OFF_RSUM   (OFF_P + (size_t)B_*ANS*ANS)       // [B][ANS]
#define OFF_CSUM2  (OFF_RSUM + (size_t)B_*ANS)        // [B][ANS]
#define OFF_ACC    (OFF_CSUM2 + (size_t)B_*ANS)       // scalar

// -------- kernel 0: start/size per (tensor,b); zero accumulators --------
__global__ void k_prep(const int* __restrict__ st, const int* __restrict__ tt,
                       float* __restrict__ ws) {
  int tid = threadIdx.x;
  int* ss = (int*)(ws + OFF_SS);
  if (tid < 2 * B_) {
    int z = tid / B_, b = tid % B_;
    const int* tg = (z == 0 ? st : tt) + b * T_;
    int start = 0, size = 0; bool found = false;
    for (int i = 0; i < T_; ++i) {
      if (tg[i] != IGNORE_) { if (!found) { start = i; found = true; } size++; }
    }
    ss[(z * B_ + b) * 2 + 0] = start;
    ss[(z * B_ + b) * 2 + 1] = size;
  }
  float* cs = ws + OFF_CSUM;
  for (int i = tid; i < 2 * V_; i += blockDim.x) cs[i] = 0.f;
  if (tid == 0) ws[OFF_ACC] = 0.f;
}

// -------- kernel 1: per-row stats (sum/sumsq/max, then sumexp) --------
__global__ void k_rowstats(const float* __restrict__ slog, const float* __restrict__ tlog,
                           float* __restrict__ ws) {
  int t = blockIdx.x, b = blockIdx.y, z = blockIdx.z;
  const int* ss = (const int*)(ws + OFF_SS);
  int start = ss[(z * B_ + b) * 2 + 0];
  int size  = ss[(z * B_ + b) * 2 + 1];
  if (t >= size) return;                       // masked row: p == 0, stats unused
  int row = start + t; row = row < 0 ? 0 : (row > T_ - 1 ? T_ - 1 : row);
  const float* x = (z == 0 ? slog : tlog) + ((size_t)b * T_ + row) * (size_t)V_;

  __shared__ float s0[256], s1[256], s2[256];
  int tid = threadIdx.x;
  float sm = 0.f, sq = 0.f, mx = -FLT_MAX;
  for (int v = tid; v < V_; v += 256) {
    __builtin_prefetch(x + v + 4096, 0, 3);
    float a = x[v];
    sm += a; sq += a * a; mx = a > mx ? a : mx;
  }
  s0[tid] = sm; s1[tid] = sq; s2[tid] = mx; __syncthreads();
  for (int o = 128; o > 0; o >>= 1) {
    if (tid < o) { s0[tid] += s0[tid + o]; s1[tid] += s1[tid + o]; s2[tid] = fmaxf(s2[tid], s2[tid + o]); }
    __syncthreads();
  }
  __shared__ float s_is, s_m;
  if (tid == 0) {
    float sum = s0[0], sumsq = s1[0], maxz = s2[0];
    float mean = sum / (float)V_;
    float var = (sumsq - (float)V_ * mean * mean) / (float)(V_ - 1);
    var = var > 0.f ? var : 0.f;
    float sd = fmaxf(sqrtf(var), 1e-6f);
    s_is = 1.f / sd;
    s_m  = maxz / sd;
  }
  __syncthreads();
  float is = s_is, m = s_m;
  float es = 0.f;
  for (int v = tid; v < V_; v += 256) es += __expf(x[v] * is - m);
  s0[tid] = es; __syncthreads();
  for (int o = 128; o > 0; o >>= 1) { if (tid < o) s0[tid] += s0[tid + o]; __syncthreads(); }
  if (tid == 0) {
    float* st3 = ws + OFF_STATS + ((size_t)(z * B_ + b) * ANS + t) * 3;
    st3[0] = is; st3[1] = m; st3[2] = 1.f / s0[0];
  }
}

// -------- kernel 2: vocab column sums of probabilities (register accumulate) --------
__global__ void k_colsum(const float* __restrict__ slog, const float* __restrict__ tlog,
                         float* __restrict__ ws) {
  int z = blockIdx.z;
  int vbase = blockIdx.x * 2048;
  int r0 = blockIdx.y * 256;                       // 2048 rows split into 8 chunks
  const int* ss = (const int*)(ws + OFF_SS);
  const float* stats = ws + OFF_STATS;
  const float* logit = (z == 0 ? slog : tlog);
  int tid = threadIdx.x;
  float acc[8];
#pragma unroll
  for (int k = 0; k < 8; ++k) acc[k] = 0.f;

  for (int rl = 0; rl < 256; ++rl) {
    int ridx = r0 + rl;
    int b = ridx / ANS, t = ridx % ANS;
    int start = ss[(z * B_ + b) * 2 + 0];
    int size  = ss[(z * B_ + b) * 2 + 1];
    if (t >= size) continue;
    int row = start + t; row = row < 0 ? 0 : (row > T_ - 1 ? T_ - 1 : row);
    const float* st3 = stats + ((size_t)(z * B_ + b) * ANS + t) * 3;
    float is = st3[0], m = st3[1], ise = st3[2];
    const float* x = logit + ((size_t)b * T_ + row) * (size_t)V_ + vbase;
#pragma unroll
    for (int k = 0; k < 8; ++k) {
      int v = tid + k * 256;
      if (vbase + v < V_) acc[k] += __expf(x[v] * is - m) * ise;
    }
  }
  float* cs = ws + OFF_CSUM + (size_t)z * V_;
#pragma unroll
  for (int k = 0; k < 8; ++k) {
    int v = vbase + tid + k * 256;
    if (v < V_) atomicAdd(&cs[v], acc[k]);
  }
}

// -------- kernel 3: stable top-50 (iterative argmax, tie -> smaller index) --------
__global__ void k_topk(float* __restrict__ ws) {
  int z = blockIdx.x, tid = threadIdx.x;
  float* cs = ws + OFF_CSUM + (size_t)z * V_;
  int* ti = (int*)(ws + OFF_TOPI) + z * TOPK;
  __shared__ float sv[256];
  __shared__ int   si[256];
  for (int i = 0; i < TOPK; ++i) {
    float bv = -FLT_MAX; int bi = V_;
    for (int v = tid; v < V_; v += 256) {
      float a = cs[v];
      if (a > bv || (a == bv && v < bi)) { bv = a; bi = v; }
    }
    sv[tid] = bv; si[tid] = bi; __syncthreads();
    for (int o = 128; o > 0; o >>= 1) {
      if (tid < o) {
        if (sv[tid + o] > sv[tid] || (sv[tid + o] == sv[tid] && si[tid + o] < si[tid])) {
          sv[tid] = sv[tid + o]; si[tid] = si[tid + o];
        }
      }
      __syncthreads();
    }
    if (tid == 0) { ti[i] = si[0]; cs[si[0]] = -FLT_MAX; }
    __syncthreads();
  }
}

// -------- kernel 4: gather top-50 probs per row, softmax(p / SINK_T) --------
__global__ void k_gather(const float* __restrict__ slog, const float* __restrict__ tlog,
                         float* __restrict__ ws) {
  int t = blockIdx.x, b = blockIdx.y, z = blockIdx.z, tid = threadIdx.x;
  const int* ss = (const int*)(ws + OFF_SS);
  int start = ss[(z * B_ + b) * 2 + 0];
  int size  = ss[(z * B_ + b) * 2 + 1];
  const int* ti = (const int*)(ws + OFF_TOPI) + z * TOPK;

  float p = 0.f;
  if (tid < TOPK && t < size) {
    int row = start + t; row = row < 0 ? 0 : (row > T_ - 1 ? T_ - 1 : row);
    const float* st3 = ws + OFF_STATS + ((size_t)(z * B_ + b) * ANS + t) * 3;
    const float* x = (z == 0 ? slog : tlog) + ((size_t)b * T_ + row) * (size_t)V_;
    p = __expf(x[ti[tid]] * st3[0] - st3[1]) * st3[2];
  }
  __shared__ float sh[64];
  float a = (tid < TOPK) ? p / SINK_T_ : -FLT_MAX;
  sh[tid] = a; __syncthreads();
  for (int o = 32; o > 0; o >>= 1) { if (tid < o) sh[tid] = fmaxf(sh[tid], sh[tid + o]); __syncthreads(); }
  float mx = sh[0]; __syncthreads();
  float e = (tid < TOPK) ? __expf(a - mx) : 0.f;
  sh[tid] = e; __syncthreads();
  for (int o = 32; o > 0; o >>= 1) { if (tid < o) sh[tid] += sh[tid + o]; __syncthreads(); }
  float inv = 1.f / sh[0];
  if (tid < TOPK)
    ws[OFF_Q + ((size_t)(z * B_ + b) * ANS + t) * TOPK + tid] = e * inv;
}

// -------- kernel 5: L1 cost matrix W and initial P = max(exp(-W/eps),1e-30) --------
__global__ void k_cost(float* __restrict__ ws) {
  int jt = blockIdx.x * 16, it = blockIdx.y * 16, b = blockIdx.z;
  __shared__ float qs[16][TOPK + 2], qt[16][TOPK + 2];
  int tid = threadIdx.y * 16 + threadIdx.x;
  const float* Qs = ws + OFF_Q + ((size_t)(0 * B_ + b) * ANS) * TOPK;
  const float* Qt = ws + OFF_Q + ((size_t)(1 * B_ + b) * ANS) * TOPK;
  for (int idx = tid; idx < 16 * TOPK; idx += 256) {
    int r = idx / TOPK, k = idx % TOPK;
    qs[r][k] = Qs[(size_t)(it + r) * TOPK + k];
    qt[r][k] = Qt[(size_t)(jt + r) * TOPK + k];
  }
  __syncthreads();
  int i = threadIdx.y, j = threadIdx.x;
  float w = 0.f;
#pragma unroll
  for (int k = 0; k < TOPK; ++k) w += fabsf(qs[i][k] - qt[j][k]);
  size_t o = ((size_t)b * ANS + it + i) * ANS + jt + j;
  ws[OFF_W + o] = w;
  ws[OFF_P + o] = fmaxf(__expf(-w / SINK_EPS_), 1e-30f);
}

// -------- Sinkhorn marginals via V_WMMA_F32_16X16X4_F32 (ones-matmul reduction) ----
// A (16x4 f32): lane m%16 holds row M; VGPR pair holds K = {0,1} (lanes 0-15)
// or {2,3} (lanes 16-31).  B (4x16): lane n%16 holds column N; VGPR pair holds
// the same K split.  C (16x16): VGPR r = row M=r (lanes 0-15) / M=r+8 (16-31).

// row sums: C = P_tile(16x4) x ones(4x16), column 0 of C = row sums
__global__ void k_rowsum(float* __restrict__ ws) {
  int b = blockIdx.y;
  int wave = threadIdx.x >> 5;
  int lane = threadIdx.x & 31;
  int rb = (blockIdx.x * 8 + wave) * 16;
  const float* P = ws + OFF_P + (size_t)b * ANS * ANS;
  int r  = rb + (lane & 15);
  int ko = (lane >> 4) * 2;
  v2f ones; ones.x = 1.f; ones.y = 1.f;
  v8f c = {};
  for (int k = 0; k < ANS; k += 4) {
    v2f a = *(const v2f*)(P + (size_t)r * ANS + k + ko);
    c = __builtin_amdgcn_wmma_f32_16x16x4_f32(false, a, false, ones, (short)0, c, false, false);
  }
  float* rs = ws + OFF_RSUM + (size_t)b * ANS;
  if (lane == 0) {
#pragma unroll
    for (int q = 0; q < 8; ++q) rs[rb + q] = c[q];
  } else if (lane == 16) {
#pragma unroll
    for (int q = 0; q < 8; ++q) rs[rb + 8 + q] = c[q];
  }
}

__global__ void k_rowdiv(float* __restrict__ ws) {
  size_t idx = (size_t)blockIdx.x * blockDim.x + threadIdx.x;
  if (idx >= (size_t)B_ * ANS * ANS) return;
  size_t b = idx / ((size_t)ANS * ANS);
  size_t i = (idx / ANS) % ANS;
  float rs = ws[OFF_RSUM + b * ANS + i];
  ws[OFF_P + idx] /= fmaxf(rs, 1e-10f);
}

// col sums: C = ones(16x4) x P_tile(4x16), row 0 of C = column sums
__global__ void k_colsumP(float* __restrict__ ws) {
  int b = blockIdx.y;
  int wave = threadIdx.x >> 5;
  int lane = threadIdx.x & 31;
  int cb = (blockIdx.x * 8 + wave) * 16;
  const float* P = ws + OFF_P + (size_t)b * ANS * ANS;
  int n  = cb + (lane & 15);
  int ko = (lane >> 4) * 2;
  v2f ones; ones.x = 1.f; ones.y = 1.f;
  v8f c = {};
  for (int k = 0; k < ANS; k += 4) {
    v2f bb;
    bb.x = P[(size_t)(k + ko) * ANS + n];
    bb.y = P[(size_t)(k + ko + 1) * ANS + n];
    c = __builtin_amdgcn_wmma_f32_16x16x4_f32(false, ones, false, bb, (short)0, c, false, false);
  }
  float* cs = ws + OFF_CSUM2 + (size_t)b * ANS;
  if (lane < 16) cs[n] = c[0];
}

__global__ void k_coldiv(float* __restrict__ ws) {
  size_t idx = (size_t)blockIdx.x * blockDim.x + threadIdx.x;
  if (idx >= (size_t)B_ * ANS * ANS) return;
  size_t b = idx / ((size_t)ANS * ANS);
  size_t j = idx % ANS;
  float cs = ws[OFF_CSUM2 + b * ANS + j];
  ws[OFF_P + idx] /= fmaxf(cs, 1e-10f);
}

// -------- final reduction: sum(P*W) --------
__global__ void k_loss(float* __restrict__ ws) {
  __shared__ float sh[256];
  size_t N = (size_t)B_ * ANS * ANS;
  float acc = 0.f;
  for (size_t i = (size_t)blockIdx.x * 256 + threadIdx.x; i < N; i += (size_t)gridDim.x * 256)
    acc += ws[OFF_P + i] * ws[OFF_W + i];
  sh[threadIdx.x] = acc; __syncthreads();
  for (int o = 128; o > 0; o >>= 1) { if (threadIdx.x < o) sh[threadIdx.x] += sh[threadIdx.x + o]; __syncthreads(); }
  if (threadIdx.x == 0) atomicAdd(&ws[OFF_ACC], sh[0]);
}

__global__ void k_final(const float* __restrict__ ws, const float* __restrict__ ce,
                        float* __restrict__ out) {
  out[0] = CE_W_ * ce[0] + KD_W_ * 0.001f * ws[OFF_ACC];
}

extern "C" void kernel_launch(void* const* d_in, const int* in_sizes, int n_in,
                              void* d_out, int out_size, void* d_ws, size_t ws_size,
                              hipStream_t stream) {
  const float* slog = (const float*)d_in[0];
  const float* tlog = (const float*)d_in[1];
  const float* ce   = (const float*)d_in[2];
  const int*   st   = (const int*)d_in[3];
  const int*   tt   = (const int*)d_in[4];
  float* ws  = (float*)d_ws;
  float* out = (float*)d_out;

  hipLaunchKernelGGL(k_prep,     dim3(1),               dim3(256),    0, stream, st, tt, ws);
  hipLaunchKernelGGL(k_rowstats, dim3(ANS, B_, 2),      dim3(256),    0, stream, slog, tlog, ws);
  hipLaunchKernelGGL(k_colsum,   dim3(16, 8, 2),        dim3(256),    0, stream, slog, tlog, ws);
  hipLaunchKernelGGL(k_topk,     dim3(2),               dim3(256),    0, stream, ws);
  hipLaunchKernelGGL(k_gather,   dim3(ANS, B_, 2),      dim3(64),     0, stream, slog, tlog, ws);
  hipLaunchKernelGGL(k_cost,     dim3(ANS/16, ANS/16, B_), dim3(16,16), 0, stream, ws);
  for (int it = 0; it < SINK_ITERS_; ++it) {
    hipLaunchKernelGGL(k_rowsum,  dim3(4, B_), dim3(256), 0, stream, ws);
    hipLaunchKernelGGL(k_rowdiv,  dim3(4096),  dim3(256), 0, stream, ws);
    hipLaunchKernelGGL(k_colsumP, dim3(4, B_), dim3(256), 0, stream, ws);
    hipLaunchKernelGGL(k_coldiv,  dim3(4096),  dim3(256), 0, stream, ws);
  }
  hipLaunchKernelGGL(k_loss,  dim3(1024), dim3(256), 0, stream, ws);
  hipLaunchKernelGGL(k_final, dim3(1),    dim3(1),   0, stream, ws, ce, out);
}